// ModulateWithZ_5171140624651
// MI455X (gfx1250) — compile-verified
//
#include <hip/hip_runtime.h>

// ---------------------------------------------------------------------------
// Fused modulated-linear kernel for MI455X (gfx1250, wave32, WMMA).
//   s      = z @ lt_w^T + lt_b                       [B, IN]
//   acc1   = (tensor * s) @ W                        [B, OUT]
//   acc2   = (s*s) @ (W*W)                           [B, OUT]
//   out    = acc1 * rsqrt(acc2 + 1e-8) + bias
//
// - V_WMMA_F32_16X16X4_F32 (exact fp32 FMA semantics, matches fp32 ref).
// - W tile read from HBM once per use; squared in-register for the demod GEMM.
// - LDS layouts tuned so every WMMA fragment is one aligned ds_load_b64:
//     A: row-major, even padded stride 34 (conflict-free, 8B aligned pairs)
//     B: K-pair interleaved float2 rows, padded stride 144 (half-wave bank split)
// - Double-buffered LDS (ping/pong), one barrier per K step: global staging of
//   tile k+1 overlaps the WMMA stream of tile k.
// ---------------------------------------------------------------------------

typedef float v2f __attribute__((ext_vector_type(2)));
typedef float v8f __attribute__((ext_vector_type(8)));

#define IN_F   1024
#define OUT_F  1024
#define LAT    16
#define BATCH  256

#define BM 32        // workgroup tile rows
#define BN 128       // workgroup tile cols
#define BK 32        // K step
#define NSTEPS (IN_F / BK)
#define NTHREADS 256

#define A_STRIDE  34   // floats; even -> 8B-aligned K pairs, conflict-free
#define WP_STRIDE 144  // float2 per pair-row; +row ≡ 32 words mod 64 banks

__global__ __launch_bounds__(NTHREADS)
void modulate_fused_kernel(const float* __restrict__ tensor,
                           const float* __restrict__ z,
                           const float* __restrict__ weight,
                           const float* __restrict__ bias,
                           const float* __restrict__ lt_w,
                           const float* __restrict__ lt_b,
                           float* __restrict__ out) {
    __shared__ float  sZ[BM * LAT];                      // z rows, loaded once
    __shared__ float  sA1[2][BM * A_STRIDE];             // (x*s) tiles
    __shared__ float  sA2[2][BM * A_STRIDE];             // (s*s) tiles
    __shared__ float2 sW1[2][(BK / 2) * WP_STRIDE];      // W, K-pair interleaved
    __shared__ float2 sW2[2][(BK / 2) * WP_STRIDE];      // W*W, same layout

    const int tid   = threadIdx.x;
    const int mBase = blockIdx.y * BM;
    const int nBase = blockIdx.x * BN;

    // Stage z rows for this M tile (32 x 16 = 2KB).
    for (int i = tid; i < BM * LAT; i += NTHREADS)
        sZ[i] = z[mBase * LAT + i];

    // Wave / lane geometry (wave32).
    const int wave  = tid >> 5;            // 0..7
    const int lane  = tid & 31;
    const int laneM = lane & 15;           // M (A frag) / N (B frag)
    const int kHalf = (lane >> 4) << 1;    // lanes 16..31 hold K+2 (ISA layout)
    const int mSub  = (wave >> 2) << 4;    // 0 or 16
    const int nSub  = (wave & 3) << 5;     // 0,32,64,96

    // Staging assignments.
    const int kkA = tid & 31;              // A: k within tile
    const int mA0 = tid >> 5;              //    m base, strided by 8
    const int pW  = tid >> 4;              // W: K-pair row 0..15
    const int chW = tid & 15;              //    n chunk

    // Stage one K-tile (k0) into buffer `buf`.
    auto stage = [&](int buf, int k0) {
        // ---- W tile + its square, K-pair interleaved -----------------------
        #pragma unroll
        for (int half = 0; half < 2; ++half) {
            const int n = (chW + half * 16) * 4;              // 0..124 step 4
            const float* g0 = weight + (size_t)(k0 + 2 * pW) * OUT_F + nBase + n;
            const float4 r0 = *(const float4*)g0;
            const float4 r1 = *(const float4*)(g0 + OUT_F);
            float2* d1 = &sW1[buf][pW * WP_STRIDE + n];
            float2* d2 = &sW2[buf][pW * WP_STRIDE + n];
            float4 lo, hi;
            lo.x = r0.x; lo.y = r1.x; lo.z = r0.y; lo.w = r1.y;
            hi.x = r0.z; hi.y = r1.z; hi.z = r0.w; hi.w = r1.w;
            *(float4*)(d1 + 0) = lo;
            *(float4*)(d1 + 2) = hi;
            float4 lo2, hi2;
            lo2.x = lo.x * lo.x; lo2.y = lo.y * lo.y;
            lo2.z = lo.z * lo.z; lo2.w = lo.w * lo.w;
            hi2.x = hi.x * hi.x; hi2.y = hi.y * hi.y;
            hi2.z = hi.z * hi.z; hi2.w = hi.w * hi.w;
            *(float4*)(d2 + 0) = lo2;
            *(float4*)(d2 + 2) = hi2;
        }
        // ---- s on the fly, stage (x*s) and (s*s) ---------------------------
        {
            const int k = k0 + kkA;
            float lw[LAT];
            #pragma unroll
            for (int c = 0; c < 4; ++c) {
                const float4 v = *(const float4*)(lt_w + (size_t)k * LAT + 4 * c);
                lw[4 * c + 0] = v.x; lw[4 * c + 1] = v.y;
                lw[4 * c + 2] = v.z; lw[4 * c + 3] = v.w;
            }
            const float lb = lt_b[k];
            #pragma unroll
            for (int i = 0; i < 4; ++i) {
                const int m = mA0 + (i << 3);
                const float* zr = &sZ[m * LAT];
                float s = lb;
                #pragma unroll
                for (int l = 0; l < LAT; ++l)
                    s += zr[l] * lw[l];
                const float x = tensor[(size_t)(mBase + m) * IN_F + k];
                sA1[buf][m * A_STRIDE + kkA] = x * s;
                sA2[buf][m * A_STRIDE + kkA] = s * s;
            }
        }
    };

    v8f acc1[2] = {};
    v8f acc2[2] = {};

    stage(0, 0);
    __syncthreads();

    for (int step = 0; step < NSTEPS; ++step) {
        const int cur = step & 1;
        if (step + 1 < NSTEPS)
            stage(cur ^ 1, (step + 1) * BK);   // overlaps with WMMA below

        // ---- WMMA over current K tile: every fragment = one ds_load_b64 ---
        #pragma unroll
        for (int kk = 0; kk < BK; kk += 4) {
            const int ar = (mSub + laneM) * A_STRIDE + kk + kHalf;
            const v2f a1 = *(const v2f*)&sA1[cur][ar];
            const v2f a2 = *(const v2f*)&sA2[cur][ar];

            const int p2 = (kk >> 1) + (lane >> 4);           // 2*p2 = kk+kHalf
            const float2* w1 = &sW1[cur][p2 * WP_STRIDE + nSub + laneM];
            const float2* w2 = &sW2[cur][p2 * WP_STRIDE + nSub + laneM];
            const v2f b10 = *(const v2f*)(w1 + 0);
            const v2f b11 = *(const v2f*)(w1 + 16);
            const v2f b20 = *(const v2f*)(w2 + 0);
            const v2f b21 = *(const v2f*)(w2 + 16);

            acc1[0] = __builtin_amdgcn_wmma_f32_16x16x4_f32(
                false, a1, false, b10, (short)0, acc1[0], false, false);
            acc2[0] = __builtin_amdgcn_wmma_f32_16x16x4_f32(
                false, a2, false, b20, (short)0, acc2[0], false, false);
            acc1[1] = __builtin_amdgcn_wmma_f32_16x16x4_f32(
                false, a1, false, b11, (short)0, acc1[1], false, false);
            acc2[1] = __builtin_amdgcn_wmma_f32_16x16x4_f32(
                false, a2, false, b21, (short)0, acc2[1], false, false);
        }
        __syncthreads();
    }

    // ---- Epilogue: out = acc1 * rsqrt(acc2 + 1e-8) + bias -----------------
    // C/D layout: VGPR r -> row (r + 8*(lane>=16)), col = lane&15.
    const int rowHalf = (lane >> 4) << 3;   // 0 or 8
    #pragma unroll
    for (int j = 0; j < 2; ++j) {
        const int col = nBase + nSub + j * 16 + laneM;
        const float bv = bias[col];
        #pragma unroll
        for (int r = 0; r < 8; ++r) {
            const int row = mBase + mSub + rowHalf + r;
            const float f = rsqrtf(acc2[j][r] + 1e-8f);
            out[(size_t)row * OUT_F + col] = acc1[j][r] * f + bv;
        }
    }
}

extern "C" void kernel_launch(void* const* d_in, const int* in_sizes, int n_in,
                              void* d_out, int out_size, void* d_ws, size_t ws_size,
                              hipStream_t stream) {
    (void)in_sizes; (void)n_in; (void)out_size; (void)d_ws; (void)ws_size;
    const float* tensor = (const float*)d_in[0];   // [256,1024]
    const float* z      = (const float*)d_in[1];   // [256,16]
    const float* weight = (const float*)d_in[2];   // [1024,1024]
    const float* bias   = (const float*)d_in[3];   // [1024]
    const float* lt_w   = (const float*)d_in[4];   // [1024,16]
    const float* lt_b   = (const float*)d_in[5];   // [1024]
    float* out = (float*)d_out;                    // [256,1024]

    dim3 grid(OUT_F / BN, BATCH / BM);             // 8 x 8 workgroups
    modulate_fused_kernel<<<grid, NTHREADS, 0, stream>>>(
        tensor, z, weight, bias, lt_w, lt_b, out);
}